// Diversity2_52699248721934
// MI455X (gfx1250) — compile-verified
//
#include <hip/hip_runtime.h>
#include <stdint.h>

// CDNA5 / gfx1250, wave32. Streaming row-wise softmax-Pearson reduction:
//   524 MB @ 23.3 TB/s -> ~22.5 us roofline.
// Data path: global_load_async_to_lds_b128 (coalesced, ASYNCcnt double-buffer)
//   -> conflict-free ds_load_b64 in WMMA A-layout
//   -> V_WMMA_F32_16X16X4_F32 row-sum accumulation (exact f32, co-executes
//      with the v_exp_f32 VALU stream).

typedef float v2f __attribute__((ext_vector_type(2)));
typedef float v8f __attribute__((ext_vector_type(8)));
typedef int   v4i __attribute__((ext_vector_type(4)));

typedef __attribute__((address_space(1))) v4i* gptr4;   // global, 16B granule
typedef __attribute__((address_space(3))) v4i* lptr4;   // LDS, 16B granule

#define TINV_LOG2E 0.07213475204444817f   // log2(e) / T, T = 20
#define SCALE_CONST 0.3f

#define CHUNK 40                 // staged columns per chunk; C = 1000 = 25 * 40
#define STRIDE 68                // LDS row stride (floats); 68 % 64 == 4 -> no bank conflicts
#define WAVES_PER_BLOCK 2
#define ROWS_PER_BLOCK (WAVES_PER_BLOCK * 16)
#define BUFSTRIDE (16u * STRIDE * 4u * WAVES_PER_BLOCK)  // bytes per double-buffer slice

// ---- async global -> LDS (ASYNCcnt path) -----------------------------------
__device__ __forceinline__ void async_copy16(const float* g, uint32_t lds_off) {
#if __has_builtin(__builtin_amdgcn_global_load_async_to_lds_b128)
  __builtin_amdgcn_global_load_async_to_lds_b128(
      (gptr4)(uintptr_t)g, (lptr4)lds_off, 0, 0);
#else
  asm volatile("global_load_async_to_lds_b128 %0, %1, off"
               :: "v"(lds_off), "v"((uint64_t)(uintptr_t)g) : "memory");
#endif
}

#if __has_builtin(__builtin_amdgcn_s_wait_asynccnt)
#define WAIT_ASYNC(n) do { __builtin_amdgcn_s_wait_asynccnt(n); \
                           asm volatile("" ::: "memory"); } while (0)
#else
#define WAIT_ASYNC(n) asm volatile("s_wait_asynccnt %0" :: "n"(n) : "memory")
#endif

__device__ __forceinline__ float v8_get(v8f v, int i) {
  switch (i & 7) {
    case 0: return v[0]; case 1: return v[1]; case 2: return v[2]; case 3: return v[3];
    case 4: return v[4]; case 5: return v[5]; case 6: return v[6];
  }
  return v[7];
}

// One wave per 16 rows, 2 waves / block. Requires C % CHUNK == 0, N % 32 == 0.
__global__ __launch_bounds__(64) void div2_rowcost_kernel(
    const float* __restrict__ x1, const float* __restrict__ x2,
    float* __restrict__ cost, int C) {
  __shared__ __align__(16) float lds1[2][ROWS_PER_BLOCK][STRIDE];
  __shared__ __align__(16) float lds2[2][ROWS_PER_BLOCK][STRIDE];

  const int lane  = threadIdx.x & 31;
  const int wave  = threadIdx.x >> 5;
  const int wrow0 = wave * 16;                                 // wave's LDS row base
  const int row0  = blockIdx.x * ROWS_PER_BLOCK + wrow0;       // wave's global row base

  // Load mapping: per matrix per chunk, a wave moves 16 rows x 40 cols = 160
  // float4 transfers = 5 async b128 instructions. Transfer f = j*32 + lane:
  // row = f/10, col = 4*(f%10)  (contiguous 160B per row -> ~full cacheline use).
  const float* g1[5]; const float* g2[5];
  uint32_t o1b[5], o2b[5];
#pragma unroll
  for (int j = 0; j < 5; ++j) {
    int f = j * 32 + lane;
    int r = f / 10, cq = f % 10;
    g1[j] = x1 + (long)(row0 + r) * C + cq * 4;
    g2[j] = x2 + (long)(row0 + r) * C + cq * 4;
    o1b[j] = (uint32_t)(uintptr_t)&lds1[0][wrow0 + r][cq * 4];
    o2b[j] = (uint32_t)(uintptr_t)&lds2[0][wrow0 + r][cq * 4];
  }

  auto issue = [&](uint32_t bufoff) {
#pragma unroll
    for (int j = 0; j < 5; ++j) async_copy16(g1[j], o1b[j] + bufoff);
#pragma unroll
    for (int j = 0; j < 5; ++j) async_copy16(g2[j], o2b[j] + bufoff);
#pragma unroll
    for (int j = 0; j < 5; ++j) { g1[j] += CHUNK; g2[j] += CHUNK; }
  };

  // WMMA A-tile (16x4 f32): lanes 0-15 -> (M=lane, K={0,1}), lanes 16-31 -> (M, K={2,3}).
  const int mrow = lane & 15;
  const int koff = (lane >> 4) << 1;

  v8f aU1 = {}, aU2 = {}, aU11 = {}, aU22 = {}, aU12 = {};
  const v2f ones = {1.0f, 1.0f};

  auto compute40 = [&](const float* b1, const float* b2) {
#pragma unroll
    for (int s = 0; s < CHUNK / 4; ++s) {
      v2f v1 = *(const v2f*)(b1 + s * 4);   // ds_load_b64, conflict-free
      v2f w2 = *(const v2f*)(b2 + s * 4);
      v2f t1, t2;                            // t = exp(x/T) - 1 (centered sums)
      t1.x = __builtin_amdgcn_exp2f(v1.x * TINV_LOG2E) - 1.0f;
      t1.y = __builtin_amdgcn_exp2f(v1.y * TINV_LOG2E) - 1.0f;
      t2.x = __builtin_amdgcn_exp2f(w2.x * TINV_LOG2E) - 1.0f;
      t2.y = __builtin_amdgcn_exp2f(w2.y * TINV_LOG2E) - 1.0f;
      v2f q11 = t1 * t1, q22 = t2 * t2, q12 = t1 * t2;
      // D = A x ones + C : exact f32 row-sum accumulation on the matrix unit.
      aU1  = __builtin_amdgcn_wmma_f32_16x16x4_f32(false, t1,  false, ones, (short)0, aU1,  false, false);
      aU2  = __builtin_amdgcn_wmma_f32_16x16x4_f32(false, t2,  false, ones, (short)0, aU2,  false, false);
      aU11 = __builtin_amdgcn_wmma_f32_16x16x4_f32(false, q11, false, ones, (short)0, aU11, false, false);
      aU22 = __builtin_amdgcn_wmma_f32_16x16x4_f32(false, q22, false, ones, (short)0, aU22, false, false);
      aU12 = __builtin_amdgcn_wmma_f32_16x16x4_f32(false, q12, false, ones, (short)0, aU12, false, false);
    }
  };

  const int nchunk = C / CHUNK;
  issue(0);                                   // prime buffer 0
  int ch = 0;
  for (;;) {
    if (ch + 1 < nchunk) { issue(BUFSTRIDE); WAIT_ASYNC(10); } else { WAIT_ASYNC(0); }
    compute40(&lds1[0][wrow0 + mrow][koff], &lds2[0][wrow0 + mrow][koff]);
    if (++ch == nchunk) break;
    if (ch + 1 < nchunk) { issue(0); WAIT_ASYNC(10); } else { WAIT_ASYNC(0); }
    compute40(&lds1[1][wrow0 + mrow][koff], &lds2[1][wrow0 + mrow][koff]);
    if (++ch == nchunk) break;
  }

  // C/D layout: VGPR g = row g (lanes 0-15) / row g+8 (lanes 16-31), all
  // columns identical. Lanes 0-7 emit rows 0-7, lanes 16-23 rows 8-15.
  const int sub = lane & 15;
  if (sub < 8) {
    const int idx  = sub;
    const int rloc = sub + ((lane >> 4) << 3);
    float u1  = v8_get(aU1,  idx);
    float u2  = v8_get(aU2,  idx);
    float u11 = v8_get(aU11, idx);
    float u22 = v8_get(aU22, idx);
    float u12 = v8_get(aU12, idx);
    float Cf  = (float)C;
    // Pearson corr of softmax rows; softmax denominators cancel exactly:
    // cost = (C*U12 - U1*U2) / sqrt((C*U11 - U1^2)(C*U22 - U2^2))
    float num = Cf * u12 - u1 * u2;
    float d1  = Cf * u11 - u1 * u1;
    float d2  = Cf * u22 - u2 * u2;
    cost[row0 + rloc] = num * __builtin_amdgcn_rsqf(d1 * d2);
  }
}

// Deterministic final mean: single block, fixed reduction tree.
__global__ __launch_bounds__(256) void div2_reduce_kernel(
    const float* __restrict__ cost, float* __restrict__ out, int N, float scale_over_n) {
  __shared__ float sm[256];
  float acc = 0.0f;
  for (int i = threadIdx.x; i < N; i += 256) acc += cost[i];
  sm[threadIdx.x] = acc;
  __syncthreads();
  for (int s = 128; s > 0; s >>= 1) {
    if ((int)threadIdx.x < s) sm[threadIdx.x] += sm[threadIdx.x + s];
    __syncthreads();
  }
  if (threadIdx.x == 0) out[0] = sm[0] * scale_over_n;
}

extern "C" void kernel_launch(void* const* d_in, const int* in_sizes, int n_in,
                              void* d_out, int out_size, void* d_ws, size_t ws_size,
                              hipStream_t stream) {
  const float* x1 = (const float*)d_in[0];   // outputs1 [N, C] f32
  const float* x2 = (const float*)d_in[1];   // outputs2 [N, C] f32
  // d_in[2] = targets (int64) — unused by the reference math.
  const int N = in_sizes[2];                 // 65536
  const int C = in_sizes[0] / N;             // 1000

  float* cost = (float*)d_ws;                // N floats = 256 KB scratch

  dim3 grid1(N / ROWS_PER_BLOCK);            // 2048 blocks of 64 threads
  div2_rowcost_kernel<<<grid1, WAVES_PER_BLOCK * 32, 0, stream>>>(x1, x2, cost, C);
  div2_reduce_kernel<<<1, 256, 0, stream>>>(cost, (float*)d_out, N,
                                            SCALE_CONST / (float)N);
}